// ChamferLoss_2173253451870
// MI455X (gfx1250) — compile-verified
//
#include <hip/hip_runtime.h>

// CDNA5 / gfx1250 Chamfer loss.
// P[m][n] = ||g_m||^2 + ||p_n||^2 - 2 g_m.p_n computed fully inside the matrix
// pipe via two chained V_WMMA_F32_16X16X4_F32 per 16x16 tile:
//   D1 = A1*B1      A1=[x,y,z,xx], B1=[-2px,-2py,-2pz,1]  -> xx - 2xy
//   P  = A2*B2 + D1 A2=[1,0,0,0],  B2=[yy,0,0,0]          -> + yy
// Epilogue is mins only (tree-structured). Each wave owns a disjoint
// 1024-column range over the block's 128 rows -> barrier-free main loop with
// software-pipelined loads; one global atomic per column per block for the
// cross-block column minima.

typedef __attribute__((ext_vector_type(2))) float v2f;
typedef __attribute__((ext_vector_type(8))) float v8f;

#define B_SZ            4
#define N_PTS           8192
#define ROWS_PER_BLOCK  128
#define COLS_PER_WAVE   (N_PTS / 8)           // 1024
#define TILES_PER_WAVE  (COLS_PER_WAVE / 16)  // 64
#define INF_KEY         0x7F800000
#define INF_F           __int_as_float(0x7F800000)

// Monotone float -> signed-int key so integer atomicMin orders floats
// correctly (including tiny negatives from rounding).
__device__ __forceinline__ int fkey(float f) {
    int i = __float_as_int(f);
    return i >= 0 ? i : (i ^ 0x7FFFFFFF);
}
__device__ __forceinline__ float keyf(int k) {
    return __int_as_float(k >= 0 ? k : (k ^ 0x7FFFFFFF));
}

__global__ void chamfer_init(int* __restrict__ colmin, float* __restrict__ loss2) {
    int i = blockIdx.x * blockDim.x + threadIdx.x;
    if (i < B_SZ * N_PTS) colmin[i] = INF_KEY;
    if (i == 0) *loss2 = 0.0f;
}

__global__ __launch_bounds__(256)
void chamfer_main(const float* __restrict__ preds, const float* __restrict__ gts,
                  int* __restrict__ colmin, float* __restrict__ loss2) {
    __shared__ int rowLDS[ROWS_PER_BLOCK];

    const int b    = blockIdx.y;
    const int row0 = blockIdx.x * ROWS_PER_BLOCK;
    const int tid  = threadIdx.x;
    const int wave = tid >> 5;
    const int lane = tid & 31;
    const int half = lane >> 4;    // 0: lanes 0-15, 1: lanes 16-31
    const int l16  = lane & 15;

    if (tid < ROWS_PER_BLOCK) rowLDS[tid] = INF_KEY;
    __syncthreads();   // rowLDS init visible before any wave's end-phase atomics

    // ---- A fragments: 8 row sub-tiles of 16 gt rows, resident in VGPRs ----
    // 16x4 f32 A layout: lanes 0-15 -> (K0,K1); lanes 16-31 -> (K2,K3)
    const float* gb = gts + (size_t)b * N_PTS * 3;
    v2f afrag[8];
#pragma unroll
    for (int rt = 0; rt < 8; ++rt) {
        const int m = row0 + rt * 16 + l16;
        const float gx = gb[(size_t)m * 3 + 0];
        const float gy = gb[(size_t)m * 3 + 1];
        const float gz = gb[(size_t)m * 3 + 2];
        const float xxm = fmaf(gx, gx, fmaf(gy, gy, gz * gz));
        afrag[rt].x = half ? gz  : gx;
        afrag[rt].y = half ? xxm : gy;
    }
    v2f a2frag;                      // [1,0,0,0] rows
    a2frag.x = half ? 0.0f : 1.0f;
    a2frag.y = 0.0f;

    float rowAcc[8][8];
#pragma unroll
    for (int rt = 0; rt < 8; ++rt)
#pragma unroll
        for (int r = 0; r < 8; ++r) rowAcc[rt][r] = INF_F;

    // ---- barrier-free main loop: this wave's disjoint column range ----
    const float* pb = preds + (size_t)b * N_PTS * 3;
    const int col0  = wave * COLS_PER_WAVE;
    int* colbase    = colmin + (size_t)b * N_PTS;

    // software pipeline: preload tile 0's pred point for this lane's column
    int n = col0 + l16;              // same column for both halves
    float cx = pb[(size_t)n * 3 + 0];
    float cy = pb[(size_t)n * 3 + 1];
    float cz = pb[(size_t)n * 3 + 2];

    for (int t = 0; t < TILES_PER_WAVE; ++t) {
        // issue next tile's load early (clamped on last iter; redundant reload)
        const int nn = (t + 1 < TILES_PER_WAVE) ? n + 16 : n;
        const float nx = pb[(size_t)nn * 3 + 0];
        const float ny = pb[(size_t)nn * 3 + 1];
        const float nz = pb[(size_t)nn * 3 + 2];
        if (t + 2 < TILES_PER_WAVE)
            __builtin_prefetch(pb + (size_t)(n + 32) * 3, 0, 0);

        const float yyn = fmaf(cx, cx, fmaf(cy, cy, cz * cz));
        v2f bfrag;                   // [-2px,-2py,-2pz,1] columns
        bfrag.x = half ? -2.0f * cz : -2.0f * cx;
        bfrag.y = half ? 1.0f       : -2.0f * cy;
        v2f b2frag;                  // [yy,0,0,0] columns
        b2frag.x = half ? 0.0f : yyn;
        b2frag.y = 0.0f;

        float rtmin[8];
#pragma unroll
        for (int rt = 0; rt < 8; ++rt) {
            v8f c = {};
            c = __builtin_amdgcn_wmma_f32_16x16x4_f32(
                false, afrag[rt], false, bfrag, (short)0, c, false, false);
            c = __builtin_amdgcn_wmma_f32_16x16x4_f32(
                false, a2frag, false, b2frag, (short)0, c, false, false);
#pragma unroll
            for (int r = 0; r < 8; ++r)
                rowAcc[rt][r] = fminf(rowAcc[rt][r], c[r]);   // row minima
            // balanced tree for this sub-tile's column minimum
            const float m0 = fminf(fminf(c[0], c[1]), fminf(c[2], c[3]));
            const float m1 = fminf(fminf(c[4], c[5]), fminf(c[6], c[7]));
            rtmin[rt] = fminf(m0, m1);
        }
        // tree over the 8 sub-tiles -> min over all 128 block rows
        float cl = fminf(fminf(fminf(rtmin[0], rtmin[1]), fminf(rtmin[2], rtmin[3])),
                         fminf(fminf(rtmin[4], rtmin[5]), fminf(rtmin[6], rtmin[7])));
        cl = fminf(cl, __shfl_xor(cl, 16, 32));
        if (lane < 16) atomicMin(&colbase[n], fkey(cl));

        cx = nx; cy = ny; cz = nz; n = nn;
    }

    // ---- row minima: lane-reduce, combine the 8 waves via LDS int keys ----
#pragma unroll
    for (int rt = 0; rt < 8; ++rt) {
#pragma unroll
        for (int r = 0; r < 8; ++r) {
            float v = rowAcc[rt][r];
            v = fminf(v, __shfl_xor(v, 1, 32));
            v = fminf(v, __shfl_xor(v, 2, 32));
            v = fminf(v, __shfl_xor(v, 4, 32));
            v = fminf(v, __shfl_xor(v, 8, 32));
            if (l16 == 0)   // lanes 0 and 16: rows r+0 / r+8 of the sub-tile
                atomicMin(&rowLDS[rt * 16 + r + 8 * half], fkey(v));
        }
    }
    __syncthreads();
    if (tid < ROWS_PER_BLOCK) {     // waves 0-3: sum the 128 row minima
        float s = keyf(rowLDS[tid]);
        s += __shfl_xor(s, 1, 32);
        s += __shfl_xor(s, 2, 32);
        s += __shfl_xor(s, 4, 32);
        s += __shfl_xor(s, 8, 32);
        s += __shfl_xor(s, 16, 32);
        if (lane == 0) atomicAdd(loss2, s);
    }
}

__global__ __launch_bounds__(256)
void chamfer_reduce(const int* __restrict__ colmin, const float* __restrict__ loss2,
                    float* __restrict__ out) {
    __shared__ float sm[256];
    float s = 0.0f;
    for (int i = threadIdx.x; i < B_SZ * N_PTS; i += 256) s += keyf(colmin[i]);
    sm[threadIdx.x] = s;
    __syncthreads();
    for (int off = 128; off > 0; off >>= 1) {
        if (threadIdx.x < off) sm[threadIdx.x] += sm[threadIdx.x + off];
        __syncthreads();
    }
    if (threadIdx.x == 0) out[0] = sm[0] + *loss2;
}

extern "C" void kernel_launch(void* const* d_in, const int* in_sizes, int n_in,
                              void* d_out, int out_size, void* d_ws, size_t ws_size,
                              hipStream_t stream) {
    (void)in_sizes; (void)n_in; (void)out_size; (void)ws_size;
    const float* preds = (const float*)d_in[0];
    const float* gts   = (const float*)d_in[1];
    float* out = (float*)d_out;

    int*   colmin = (int*)d_ws;                                   // B*N int keys
    float* loss2  = (float*)((char*)d_ws +
                             (size_t)B_SZ * N_PTS * sizeof(int)); // 1 float

    chamfer_init<<<(B_SZ * N_PTS + 255) / 256, 256, 0, stream>>>(colmin, loss2);

    dim3 grid(N_PTS / ROWS_PER_BLOCK, B_SZ);
    chamfer_main<<<grid, 256, 0, stream>>>(preds, gts, colmin, loss2);

    chamfer_reduce<<<1, 256, 0, stream>>>(colmin, loss2, out);
}